// GCN_58506044506595
// MI455X (gfx1250) — compile-verified
//
#include <hip/hip_runtime.h>

typedef __attribute__((ext_vector_type(2))) float v2f;
typedef __attribute__((ext_vector_type(8))) float v8f;

#define N_NODES 100000
#define F_INF   128
#define H_DIM   64
#define C_DIM   40
#define M_TILES (N_NODES / 16)   // 6250, exact

// ---------------- degree / normalization prep ----------------

__global__ void k_deg_init(float* __restrict__ deg) {
  int i = blockIdx.x * blockDim.x + threadIdx.x;
  if (i < N_NODES) deg[i] = 1.0f;              // self-loop weight 1
}

__global__ void k_deg_accum(const int* __restrict__ dst, const float* __restrict__ ew,
                            float* __restrict__ deg, int E) {
  int e = blockIdx.x * blockDim.x + threadIdx.x;
  if (e < E) atomicAdd(&deg[dst[e]], ew[e]);
}

__global__ void k_dinv(float* __restrict__ deg) {
  int i = blockIdx.x * blockDim.x + threadIdx.x;
  if (i < N_NODES) {
    float d = deg[i];
    deg[i] = (d > 0.0f) ? rsqrtf(d) : 0.0f;    // deg >= 1 always, but match reference
  }
}

__global__ void k_normw(const int* __restrict__ src, const int* __restrict__ dst,
                        const float* __restrict__ ew, const float* __restrict__ dinv,
                        float* __restrict__ nw, int E) {
  int e = blockIdx.x * blockDim.x + threadIdx.x;
  if (e < E) nw[e] = dinv[src[e]] * ew[e] * dinv[dst[e]];
}

// ---------------- WMMA GEMM 1: h1 = x @ W1  (K=128, Nout=64) ----------------
// One wave per 16-row stripe, computing all 4 column tiles (16x64) so each
// A fragment load feeds 4 v_wmma ops.

__global__ void k_gemm1_wmma(const float* __restrict__ x, const float* __restrict__ W1,
                             float* __restrict__ h1) {
  const int lane    = threadIdx.x & 31;
  const int mtile   = blockIdx.x * (blockDim.x >> 5) + (threadIdx.x >> 5); // wave id
  if (mtile >= M_TILES) return;
  const int m0      = mtile * 16;
  const int halfSel = lane >> 4;         // 0: K pair {0,1}, 1: K pair {2,3}
  const int lr      = lane & 15;

  v8f c0 = {}, c1 = {}, c2 = {}, c3 = {};
  const float* xrow = x + (size_t)(m0 + lr) * F_INF;
  #pragma unroll 2
  for (int kk = 0; kk < F_INF; kk += 4) {
    const int ka = kk + halfSel * 2;
    v2f a; a.x = xrow[ka]; a.y = xrow[ka + 1];
    const float* wrow = W1 + ka * H_DIM + lr;
    v2f b0; b0.x = wrow[0];  b0.y = wrow[H_DIM + 0];
    v2f b1; b1.x = wrow[16]; b1.y = wrow[H_DIM + 16];
    v2f b2; b2.x = wrow[32]; b2.y = wrow[H_DIM + 32];
    v2f b3; b3.x = wrow[48]; b3.y = wrow[H_DIM + 48];
    c0 = __builtin_amdgcn_wmma_f32_16x16x4_f32(false, a, false, b0, (short)0, c0, false, false);
    c1 = __builtin_amdgcn_wmma_f32_16x16x4_f32(false, a, false, b1, (short)0, c1, false, false);
    c2 = __builtin_amdgcn_wmma_f32_16x16x4_f32(false, a, false, b2, (short)0, c2, false, false);
    c3 = __builtin_amdgcn_wmma_f32_16x16x4_f32(false, a, false, b3, (short)0, c3, false, false);
  }
  const int rowBase = m0 + halfSel * 8;
  #pragma unroll
  for (int r = 0; r < 8; ++r) {
    float* orow = h1 + (size_t)(rowBase + r) * H_DIM + lr;
    orow[0]  = c0[r];
    orow[16] = c1[r];
    orow[32] = c2[r];
    orow[48] = c3[r];
  }
}

// ---------------- WMMA GEMM 2: h2 = relu(agg1 + b1) @ W2 (K=64, Nout=40->48) ----------------

__global__ void k_gemm2_wmma(const float* __restrict__ agg1, const float* __restrict__ b1,
                             const float* __restrict__ W2, float* __restrict__ h2) {
  const int lane    = threadIdx.x & 31;
  const int mtile   = blockIdx.x * (blockDim.x >> 5) + (threadIdx.x >> 5);
  if (mtile >= M_TILES) return;
  const int m0      = mtile * 16;
  const int halfSel = lane >> 4;
  const int lr      = lane & 15;
  const bool tailOk = (32 + lr) < C_DIM;   // third tile covers cols 32..47, valid < 40

  v8f c0 = {}, c1 = {}, c2 = {};
  const float* arow = agg1 + (size_t)(m0 + lr) * H_DIM;
  #pragma unroll 2
  for (int kk = 0; kk < H_DIM; kk += 4) {
    const int ka = kk + halfSel * 2;
    v2f a;
    a.x = fmaxf(arow[ka]     + b1[ka],     0.0f);   // fused bias + ReLU
    a.y = fmaxf(arow[ka + 1] + b1[ka + 1], 0.0f);
    const float* wrow = W2 + ka * C_DIM + lr;
    v2f b0; b0.x = wrow[0];  b0.y = wrow[C_DIM + 0];
    v2f b1v; b1v.x = wrow[16]; b1v.y = wrow[C_DIM + 16];
    v2f b2;
    b2.x = tailOk ? wrow[32]         : 0.0f;
    b2.y = tailOk ? wrow[C_DIM + 32] : 0.0f;
    c0 = __builtin_amdgcn_wmma_f32_16x16x4_f32(false, a, false, b0,  (short)0, c0, false, false);
    c1 = __builtin_amdgcn_wmma_f32_16x16x4_f32(false, a, false, b1v, (short)0, c1, false, false);
    c2 = __builtin_amdgcn_wmma_f32_16x16x4_f32(false, a, false, b2,  (short)0, c2, false, false);
  }
  const int rowBase = m0 + halfSel * 8;
  #pragma unroll
  for (int r = 0; r < 8; ++r) {
    float* orow = h2 + (size_t)(rowBase + r) * C_DIM + lr;
    orow[0]  = c0[r];
    orow[16] = c1[r];
    if (tailOk) orow[32] = c2[r];
  }
}

// ---------------- aggregation: self-loop init + edge scatter ----------------

// agg[i*F + f] = dinv[i]^2 * h[i*F + f]   (self-loop message, norm = dinv*1*dinv)
__global__ void k_agg_init(const float* __restrict__ dinv, const float* __restrict__ h,
                           float* __restrict__ agg, int F, int total) {
  int i = blockIdx.x * blockDim.x + threadIdx.x;
  if (i >= total) return;
  int node = i / F;
  float s = dinv[node];
  agg[i] = s * s * h[i];
}

// one wave per edge; lane covers features f, f+32
__global__ void k_agg_edges(const int* __restrict__ src, const int* __restrict__ dst,
                            const float* __restrict__ nw, const float* __restrict__ h,
                            float* __restrict__ agg, int F, int E) {
  const int wid  = (blockIdx.x * blockDim.x + threadIdx.x) >> 5;
  const int lane = threadIdx.x & 31;
  if (wid >= E) return;
  const int   s = src[wid];
  const int   d = dst[wid];
  const float w = nw[wid];
  const float* hs = h   + (size_t)s * F;
  float*       ad = agg + (size_t)d * F;
  #pragma unroll 2
  for (int f = lane; f < F; f += 32)
    atomicAdd(&ad[f], w * hs[f]);
}

// ---------------- final: add bias, L2-normalize each row of d_out in place ----------------

__global__ void k_finalize(float* __restrict__ out, const float* __restrict__ b2) {
  const int row  = (blockIdx.x * blockDim.x + threadIdx.x) >> 5;
  const int lane = threadIdx.x & 31;
  if (row >= N_NODES) return;
  float* o = out + (size_t)row * C_DIM;
  float v0 = o[lane] + b2[lane];                              // lane < 32 < 40
  float v1 = (lane < C_DIM - 32) ? (o[lane + 32] + b2[lane + 32]) : 0.0f;
  float ss = v0 * v0 + v1 * v1;
  #pragma unroll
  for (int m = 16; m >= 1; m >>= 1) ss += __shfl_xor(ss, m, 32);
  const float sc = 1.0f / fmaxf(sqrtf(ss), 1e-12f);
  o[lane] = v0 * sc;
  if (lane < C_DIM - 32) o[lane + 32] = v1 * sc;
}

// ---------------- launch ----------------

extern "C" void kernel_launch(void* const* d_in, const int* in_sizes, int n_in,
                              void* d_out, int out_size, void* d_ws, size_t ws_size,
                              hipStream_t stream) {
  const float* x   = (const float*)d_in[0];
  const int*   ei  = (const int*)d_in[1];     // [2, E] flat: src then dst
  const float* ew  = (const float*)d_in[2];
  const float* W1  = (const float*)d_in[3];
  const float* b1  = (const float*)d_in[4];
  const float* W2  = (const float*)d_in[5];
  const float* b2  = (const float*)d_in[6];
  float* out = (float*)d_out;

  const int E = in_sizes[2];                  // 1,600,000
  const int* src = ei;
  const int* dst = ei + E;

  // workspace layout (floats)
  float* ws   = (float*)d_ws;
  float* dinv = ws;                                    // N         (deg -> dinv in place)
  float* nw   = dinv + N_NODES;                        // E
  float* h1   = nw   + E;                              // N*64
  float* agg1 = h1   + (size_t)N_NODES * H_DIM;        // N*64
  float* h2   = agg1 + (size_t)N_NODES * H_DIM;        // N*40

  const int TB = 256;
  const int WPB = TB >> 5;                    // waves per block

  // 1) degree (init to self-loop weight, scatter edge weights, rsqrt)
  k_deg_init <<<(N_NODES + TB - 1) / TB, TB, 0, stream>>>(dinv);
  k_deg_accum<<<(E + TB - 1) / TB, TB, 0, stream>>>(dst, ew, dinv, E);
  k_dinv     <<<(N_NODES + TB - 1) / TB, TB, 0, stream>>>(dinv);
  k_normw    <<<(E + TB - 1) / TB, TB, 0, stream>>>(src, dst, ew, dinv, nw, E);

  // 2) layer 1: WMMA GEMM (wave computes 16x64), self-loop init, edge scatter
  k_gemm1_wmma<<<(M_TILES + WPB - 1) / WPB, TB, 0, stream>>>(x, W1, h1);
  k_agg_init<<<((N_NODES * H_DIM) + TB - 1) / TB, TB, 0, stream>>>(
      dinv, h1, agg1, H_DIM, N_NODES * H_DIM);
  k_agg_edges<<<(E + WPB - 1) / WPB, TB, 0, stream>>>(src, dst, nw, h1, agg1, H_DIM, E);

  // 3) layer 2: WMMA GEMM (wave computes 16x48, bias+relu fused), init into d_out, scatter
  k_gemm2_wmma<<<(M_TILES + WPB - 1) / WPB, TB, 0, stream>>>(agg1, b1, W2, h2);
  k_agg_init<<<((N_NODES * C_DIM) + TB - 1) / TB, TB, 0, stream>>>(
      dinv, h2, out, C_DIM, N_NODES * C_DIM);
  k_agg_edges<<<(E + WPB - 1) / WPB, TB, 0, stream>>>(src, dst, nw, h2, out, C_DIM, E);

  // 4) bias + row L2 normalization (in place on d_out)
  k_finalize<<<(N_NODES + WPB - 1) / WPB, TB, 0, stream>>>(out, b2);
}